// MultiHeadAttention_45122926412465
// MI455X (gfx1250) — compile-verified
//
#include <hip/hip_runtime.h>

// Problem constants (from reference)
#define H   16
#define D   64
#define SL  1024
#define BS  64
#define DM  1024
#define HD  (H * D)   // 1024

typedef float v2f __attribute__((ext_vector_type(2)));
typedef float v4f __attribute__((ext_vector_type(4)));
typedef float v8f __attribute__((ext_vector_type(8)));

// ---------------------------------------------------------------------------
// Stage 1a: qp[b, hd] = sum_dm query[b,dm] * Wq[hd,dm]     (64x1024 output)
// ---------------------------------------------------------------------------
__global__ void k_proj_qp(const float* __restrict__ query,
                          const float* __restrict__ Wq,
                          float* __restrict__ qp) {
  const int b = blockIdx.x;          // 0..63
  const int t = threadIdx.x;         // 0..255
  const float* qrow = query + (size_t)b * DM;
#pragma unroll
  for (int i = 0; i < 4; ++i) {
    const int hd = t + i * 256;
    const float* wrow = Wq + (size_t)hd * DM;
    float acc = 0.f;
    for (int dm = 0; dm < DM; dm += 4) {
      v4f w = *(const v4f*)(wrow + dm);
      v4f q = *(const v4f*)(qrow + dm);
      acc = fmaf(w.x, q.x, acc);
      acc = fmaf(w.y, q.y, acc);
      acc = fmaf(w.z, q.z, acc);
      acc = fmaf(w.w, q.w, acc);
    }
    qp[(size_t)b * HD + hd] = acc;
  }
}

// ---------------------------------------------------------------------------
// Stage 1b: qk[b,h,dm] = sum_d qp[b,h,d] * Wk[h*64+d, dm]  (64x16x1024)
// qp operand is block-uniform -> scalar loads; Wk loads coalesced over dm.
// ---------------------------------------------------------------------------
__global__ void k_proj_qk(const float* __restrict__ qp,
                          const float* __restrict__ Wk,
                          float* __restrict__ qk) {
  const int h = blockIdx.x;          // 0..15
  const int b = blockIdx.y;          // 0..63
  const int t = threadIdx.x;         // 0..255
  const float* qph = qp + (size_t)b * HD + h * D;   // 64 uniform scalars
  float acc0 = 0.f, acc1 = 0.f, acc2 = 0.f, acc3 = 0.f;
  for (int d = 0; d < D; ++d) {
    const float qv = qph[d];
    const float* wrow = Wk + ((size_t)h * D + d) * DM;
    acc0 = fmaf(qv, wrow[t + 0 * 256], acc0);
    acc1 = fmaf(qv, wrow[t + 1 * 256], acc1);
    acc2 = fmaf(qv, wrow[t + 2 * 256], acc2);
    acc3 = fmaf(qv, wrow[t + 3 * 256], acc3);
  }
  float* o = qk + ((size_t)b * H + h) * DM;
  o[t + 0 * 256] = acc0;
  o[t + 1 * 256] = acc1;
  o[t + 2 * 256] = acc2;
  o[t + 3 * 256] = acc3;
}

// ---------------------------------------------------------------------------
// Stage 2 (WMMA, reads keys once = 256 MB):
//   scores[s,b,h] = (1/8) * sum_dm keys[s,b,dm] * qk[b,h,dm]
// One wave per (s-tile, b): C[16s x 16h] += A[16s x 4k] * B[4k x 16h],
// K loop over DM in steps of 4 using v_wmma_f32_16x16x4_f32.
// A: lane l -> row M=l&15, K pair at 2*(l>>4)  (float2, contiguous in dm)
// B: lane l -> col N=l&15 (=h), K pair at 2*(l>>4) (float2 from qk row)
// ---------------------------------------------------------------------------
__global__ void k_scores_wmma(const float* __restrict__ keys,
                              const float* __restrict__ qk,
                              float* __restrict__ scores) {
  const int wave = threadIdx.x >> 5;             // 0..3
  const int lane = threadIdx.x & 31;
  const int job  = blockIdx.x * 4 + wave;        // 0..4095
  const int st   = job & 63;                     // s-tile index
  const int b    = job >> 6;                     // batch
  const int row  = lane & 15;                    // M for A, N(=h) for B/D
  const int kh   = lane >> 4;                    // K half select

  const float* ap = keys + ((size_t)(st * 16 + row)) * (BS * DM)
                         + (size_t)b * DM + 2 * kh;
  const float* bp = qk + ((size_t)b * H + row) * DM + 2 * kh;

  v8f c = {0.f, 0.f, 0.f, 0.f, 0.f, 0.f, 0.f, 0.f};
#pragma unroll 8
  for (int k = 0; k < DM; k += 4) {
    v2f a  = *(const v2f*)(ap + k);
    v2f bb = *(const v2f*)(bp + k);
    c = __builtin_amdgcn_wmma_f32_16x16x4_f32(false, a, false, bb,
                                              (short)0, c, false, false);
  }
  // D layout: VGPR r, lanes 0-15 -> M=r, lanes 16-31 -> M=r+8; N = lane&15
  float* o = scores + ((size_t)(st * 16 + 8 * kh)) * (BS * H)
                    + (size_t)b * H + row;
#pragma unroll
  for (int r = 0; r < 8; ++r)
    o[(size_t)r * (BS * H)] = c[r] * 0.125f;     // 1/sqrt(64)
}

// ---------------------------------------------------------------------------
// Stage 3: softmax over s (in place). One block per (b,h) column.
// ---------------------------------------------------------------------------
__global__ void k_softmax(float* __restrict__ sc) {
  const int col = blockIdx.x;        // b*H + h, 0..1023
  const int t   = threadIdx.x;       // 0..255
  __shared__ float red[256];

  float v[4];
#pragma unroll
  for (int i = 0; i < 4; ++i)
    v[i] = sc[(size_t)(t + i * 256) * (BS * H) + col];

  float m = fmaxf(fmaxf(v[0], v[1]), fmaxf(v[2], v[3]));
  red[t] = m;
  __syncthreads();
  for (int off = 128; off > 0; off >>= 1) {
    if (t < off) red[t] = fmaxf(red[t], red[t + off]);
    __syncthreads();
  }
  m = red[0];
  __syncthreads();

  float s = 0.f;
#pragma unroll
  for (int i = 0; i < 4; ++i) {
    v[i] = __expf(v[i] - m);
    s += v[i];
  }
  red[t] = s;
  __syncthreads();
  for (int off = 128; off > 0; off >>= 1) {
    if (t < off) red[t] += red[t + off];
    __syncthreads();
  }
  const float inv = 1.f / red[0];
#pragma unroll
  for (int i = 0; i < 4; ++i)
    sc[(size_t)(t + i * 256) * (BS * H) + col] = v[i] * inv;
}

// ---------------------------------------------------------------------------
// Stage 4 (WMMA, reads values once = 256 MB):
//   wv[b,h,dm] = sum_s attn[s,b,h] * values[s,b,dm]
// One wave per (dm-tile, b): C[16h x 16dm] += A[16h x 4s] * B[4s x 16dm].
// ---------------------------------------------------------------------------
__global__ void k_wv_wmma(const float* __restrict__ attn,
                          const float* __restrict__ values,
                          float* __restrict__ wv) {
  const int wave = threadIdx.x >> 5;
  const int lane = threadIdx.x & 31;
  const int job  = blockIdx.x * 4 + wave;        // 0..4095
  const int nt   = job & 63;                     // dm-tile index
  const int b    = job >> 6;
  const int row  = lane & 15;                    // M(=h) for A, N(=dm) for B/D
  const int kh   = lane >> 4;

  // A[h][s] = attn[s,b,h]; lane needs s = k+2*kh, k+2*kh+1 (stride BS*H)
  const float* ap = attn + (size_t)(2 * kh) * (BS * H) + (size_t)b * H + row;
  // B[s][n] = values[s,b,nt*16+n]; stride BS*DM between s
  const float* bp = values + (size_t)(2 * kh) * (BS * DM)
                           + (size_t)b * DM + nt * 16 + row;

  v8f c = {0.f, 0.f, 0.f, 0.f, 0.f, 0.f, 0.f, 0.f};
#pragma unroll 4
  for (int k = 0; k < SL; k += 4) {
    v2f a, bb;
    a.x  = ap[(size_t)(k)     * (BS * H)];
    a.y  = ap[(size_t)(k + 1) * (BS * H)];
    bb.x = bp[(size_t)(k)     * (BS * DM)];
    bb.y = bp[(size_t)(k + 1) * (BS * DM)];
    c = __builtin_amdgcn_wmma_f32_16x16x4_f32(false, a, false, bb,
                                              (short)0, c, false, false);
  }
  // D: M = r + 8*kh (= h), N = row (= dm within tile)
  float* o = wv + ((size_t)b * H + 8 * kh) * DM + nt * 16 + row;
#pragma unroll
  for (int r = 0; r < 8; ++r)
    o[(size_t)r * DM] = c[r];
}

// ---------------------------------------------------------------------------
// Stage 5: out[b, hd] = sum_dm wv[b, hd>>6, dm] * Wk[hd, dm]
// ---------------------------------------------------------------------------
__global__ void k_proj_out(const float* __restrict__ wv,
                           const float* __restrict__ Wk,
                           float* __restrict__ out) {
  const int b = blockIdx.x;
  const int t = threadIdx.x;
#pragma unroll
  for (int i = 0; i < 4; ++i) {
    const int hd = t + i * 256;
    const int h  = hd >> 6;
    const float* wrow = Wk + (size_t)hd * DM;
    const float* vrow = wv + ((size_t)b * H + h) * DM;
    float acc = 0.f;
    for (int dm = 0; dm < DM; dm += 4) {
      v4f w = *(const v4f*)(wrow + dm);
      v4f x = *(const v4f*)(vrow + dm);
      acc = fmaf(w.x, x.x, acc);
      acc = fmaf(w.y, x.y, acc);
      acc = fmaf(w.z, x.z, acc);
      acc = fmaf(w.w, x.w, acc);
    }
    out[(size_t)b * HD + hd] = acc;
  }
}

// ---------------------------------------------------------------------------
extern "C" void kernel_launch(void* const* d_in, const int* in_sizes, int n_in,
                              void* d_out, int out_size, void* d_ws, size_t ws_size,
                              hipStream_t stream) {
  const float* query  = (const float*)d_in[0];   // [BS, DM]
  const float* keys   = (const float*)d_in[1];   // [SL, BS, DM]
  const float* values = (const float*)d_in[2];   // [SL, BS, DM]
  const float* Wq     = (const float*)d_in[3];   // [HD, DM]
  const float* Wk     = (const float*)d_in[4];   // [HD, DM]
  float* out = (float*)d_out;                    // [BS, HD]

  // Workspace carve-up (floats): qp 64K | qk 1M | scores 1M | wv 1M  (~12.3 MB)
  float* qp = (float*)d_ws;
  float* qk = qp + (size_t)BS * HD;
  float* sc = qk + (size_t)BS * H * DM;
  float* wv = sc + (size_t)SL * BS * H;

  k_proj_qp  <<<BS, 256, 0, stream>>>(query, Wq, qp);
  k_proj_qk  <<<dim3(H, BS), 256, 0, stream>>>(qp, Wk, qk);
  k_scores_wmma<<<(64 * BS) / 4, 128, 0, stream>>>(keys, qk, sc);
  k_softmax  <<<BS * H, 256, 0, stream>>>(sc);
  k_wv_wmma  <<<(64 * BS) / 4, 128, 0, stream>>>(sc, values, wv);
  k_proj_out <<<BS, 256, 0, stream>>>(wv, Wk, out);
}